// Attn_82832739270904
// MI455X (gfx1250) — compile-verified
//
#include <hip/hip_runtime.h>
#include <math.h>

// ---------------- problem constants ----------------
#define BATCH   16
#define LEN     1024
#define DIM_IN  256
#define PAST    16
#define LP      (LEN - PAST)     // 1008
#define OUTF    256              // FUTURE*DIM
#define NLAYERS 3

// attention-layer kernel tiling
#define TILE    64
#define HALO    8                // need 7 halo rows (1+2+4), round to 8
#define BROWS   (TILE + HALO)    // 72 rows per LDS buffer

// projection kernel tiling
#define PROJ_ROWS 64
#define KPANEL    64
#define XSTRIDE   260            // 260 mod 64 = 4 -> conflict-free column reads
#define WSTRIDE   68             // 68 mod 64 = 4; 272 B rows stay 16B-aligned

typedef float v2f __attribute__((ext_vector_type(2)));
typedef float v4f __attribute__((ext_vector_type(4)));
typedef float v8f __attribute__((ext_vector_type(8)));
typedef int   v4i __attribute__((ext_vector_type(4)));

typedef void __attribute__((address_space(1))) *gas_ptr;
typedef void __attribute__((address_space(3))) *las_ptr;
typedef v4i  __attribute__((address_space(1))) *gv4i_ptr;
typedef v4i  __attribute__((address_space(3))) *lv4i_ptr;

#if defined(__has_builtin)
#if __has_builtin(__builtin_amdgcn_global_load_async_to_lds_b128) && \
    __has_builtin(__builtin_amdgcn_s_wait_asynccnt)
#define USE_ASYNC_LDS 1
#endif
#endif

__device__ __forceinline__ float dot4(float4 a, float4 b) {
  return a.x * b.x + a.y * b.y + a.z * b.z + a.w * b.w;
}

// =====================================================================
// Kernel 1: 3 fused dilated-attention layers.
//   - one block per (64-row tile, batch); 8 wave32 waves; wave-per-row
//   - double-buffered LDS tiles (72 rows x 256 f32 each = 147 KB total)
//   - initial tile staged with global_load_async_to_lds_b128 (ASYNCcnt)
//   - writes the dense (mostly-zero) attn rows with NT stores (201 MB
//     stream must not thrash the 192 MB L2 that keeps out3 resident)
//   - writes layer-3 output rows (out3) to workspace (L2-resident, RT)
// =====================================================================
__global__ __launch_bounds__(256) void attn_layers_kernel(
    const float* __restrict__ inp, float* __restrict__ attn_out,
    float* __restrict__ out3)
{
  extern __shared__ float smem[];
  float* bufA = smem;                       // BROWS * DIM_IN
  float* bufB = smem + BROWS * DIM_IN;

  const int t0   = blockIdx.x * TILE;
  const int b    = blockIdx.y;
  const int base = t0 - HALO;               // global row of LDS buffer row 0

  const int tid  = threadIdx.x;
  const int lane = tid & 31;
  const int wave = tid >> 5;

  // ---- stage inp rows [t0-7, t0+TILE) into bufA rows [1, 72) ----
  {
    const float4* src = (const float4*)(inp + (size_t)b * LEN * DIM_IN);
    for (int idx = tid; idx < (BROWS - 1) * (DIM_IN / 4); idx += 256) {
      const int r = idx >> 6;               // DIM_IN/4 == 64
      const int c = idx & 63;
      const int l = base + 1 + r;
      if (l >= 0) {
        const float4* gp = src + (size_t)l * 64 + c;
        float4*       lp = (float4*)(bufA + (size_t)(1 + r) * DIM_IN) + c;
#ifdef USE_ASYNC_LDS
        __builtin_amdgcn_global_load_async_to_lds_b128(
            (gv4i_ptr)(gas_ptr)(void*)const_cast<float4*>(gp),
            (lv4i_ptr)(las_ptr)(void*)lp, 0, 0);
#else
        *lp = *gp;
#endif
      }
    }
#ifdef USE_ASYNC_LDS
    __builtin_amdgcn_s_wait_asynccnt(0);
#endif
  }
  __syncthreads();

  float* cur = bufA;
  float* nxt = bufB;

#pragma unroll
  for (int layer = 0; layer < NLAYERS; ++layer) {
    const int d  = 1 << layer;                       // dilation 1,2,4
    const int lo = (layer == 0) ? t0 - 6 : (layer == 1) ? t0 - 4 : t0;
    const int hi = t0 + TILE;
    const bool last = (layer == NLAYERS - 1);

    for (int l = lo + wave; l < hi; l += 8) {
      if (l < 0) continue;
      const float4* rowL = (const float4*)(cur + (size_t)(l - base) * DIM_IN);
      const float4 x0 = rowL[lane];
      const float4 x1 = rowL[lane + 32];
      const bool hasPrev = (l - d) >= 0;
      float4 y0 = make_float4(0.f, 0.f, 0.f, 0.f);
      float4 y1 = make_float4(0.f, 0.f, 0.f, 0.f);
      if (hasPrev) {
        const float4* rowM =
            (const float4*)(cur + (size_t)(l - d - base) * DIM_IN);
        y0 = rowM[lane];
        y1 = rowM[lane + 32];
      }
      float s0 = dot4(x0, x0) + dot4(x1, x1);
      float s1 = dot4(x0, y0) + dot4(x1, y1);
#pragma unroll
      for (int off = 16; off > 0; off >>= 1) {       // wave32 reduction
        s0 += __shfl_xor(s0, off, 32);
        s1 += __shfl_xor(s1, off, 32);
      }
      s0 *= 0.0625f;                                 // 1/sqrt(256)
      s1 *= 0.0625f;
      const float mx = hasPrev ? fmaxf(s0, s1) : s0;
      const float e0 = __expf(s0 - mx);
      const float e1 = hasPrev ? __expf(s1 - mx) : 0.0f;
      const float inv = 1.0f / (e0 + e1);
      const float a0 = e0 * inv;
      const float a1 = e1 * inv;

      // out'[l] = (1+a0)*x + a1*y
      const float c0 = 1.0f + a0;
      float4 z0, z1;
      z0.x = c0 * x0.x + a1 * y0.x;  z0.y = c0 * x0.y + a1 * y0.y;
      z0.z = c0 * x0.z + a1 * y0.z;  z0.w = c0 * x0.w + a1 * y0.w;
      z1.x = c0 * x1.x + a1 * y1.x;  z1.y = c0 * x1.y + a1 * y1.y;
      z1.z = c0 * x1.z + a1 * y1.z;  z1.w = c0 * x1.w + a1 * y1.w;
      if (!last) {
        float4* dst = (float4*)(nxt + (size_t)(l - base) * DIM_IN);
        dst[lane]      = z0;
        dst[lane + 32] = z1;
      } else {
        float4* dst = (float4*)(out3 + ((size_t)b * LEN + l) * DIM_IN);
        dst[lane]      = z0;
        dst[lane + 32] = z1;
      }

      // dense attn row (zeros except cols l and l-d) for owned rows only
      if (l >= t0) {
        v4f* arow = (v4f*)(attn_out +
            (((size_t)b * NLAYERS + layer) * LEN + l) * LEN);
        const int pm = hasPrev ? (l - d) : -1;
#pragma unroll
        for (int j = 0; j < 8; ++j) {
          const int cbase = j * 128 + lane * 4;
          float v[4] = {0.f, 0.f, 0.f, 0.f};
          const int rl = l - cbase;
          if (rl >= 0 && rl < 4) v[rl] = a0;
          const int rm = pm - cbase;
          if (rm >= 0 && rm < 4) v[rm] = a1;
          v4f vv = {v[0], v[1], v[2], v[3]};
          __builtin_nontemporal_store(vv, arow + j * 32 + lane);
        }
      }
    }
    __syncthreads();
    float* t = cur; cur = nxt; nxt = t;
  }
}

// =====================================================================
// Kernel 2: o = out3[:,16:,:] @ W^T + b  (+ residual), via WMMA fp32.
//   - block: 64 rows x 256 cols; 8 waves, each owns a 16-row strip and
//     8 of the 16 column tiles -> 8 x v8f accumulators per wave
//   - X tile in LDS (stride 260); W panel staged [n][k] with stride 68
//     so both A and B operands load as conflict-free ds_load_b64
//   - v_wmma_f32_16x16x4_f32, K looped 4 at a time
// =====================================================================
__global__ __launch_bounds__(256) void proj_kernel(
    const float* __restrict__ x,        // out3 (B, LEN, 256)
    const float* __restrict__ W,        // (256, 256) row-major [n][k]
    const float* __restrict__ bias,     // (256,)
    const float* __restrict__ inp,      // residual source (B, LEN, 256)
    float* __restrict__ o)              // (B, LP, 256)
{
  extern __shared__ float smem[];
  float* Xs = smem;                         // PROJ_ROWS x XSTRIDE
  float* Ws = smem + PROJ_ROWS * XSTRIDE;   // OUTF x WSTRIDE  (W panel [n][k])

  const int tid  = threadIdx.x;
  const int lane = tid & 31;
  const int wave = tid >> 5;
  const int m0   = blockIdx.x * PROJ_ROWS;

  // ---- stage X tile ----
  for (int idx = tid; idx < PROJ_ROWS * (DIM_IN / 4); idx += 256) {
    const int r = idx >> 6;
    const int c = idx & 63;
    const int m  = m0 + r;
    const int bb = m / LP;
    const int lp = m - bb * LP;
    const float4 v =
        ((const float4*)x)[((size_t)bb * LEN + lp + PAST) * 64 + c];
    ((float4*)(Xs + (size_t)r * XSTRIDE))[c] = v;
  }

  const int mt     = wave & 3;         // 16-row strip within tile
  const int ntBase = (wave >> 2) * 8;  // first of 8 column tiles
  const int lm     = lane & 15;
  const int half   = lane >> 4;        // k sub-pair selector (A/B layout)

  v8f acc[8];
  const v8f vzero = {0.f, 0.f, 0.f, 0.f, 0.f, 0.f, 0.f, 0.f};
#pragma unroll
  for (int i = 0; i < 8; ++i) acc[i] = vzero;

  for (int kp = 0; kp < DIM_IN; kp += KPANEL) {
    __syncthreads();                    // X ready / previous panel consumed
    // stage W panel, row-major along k: Ws[n][k] = W[n][kp+k]
    for (int idx = tid; idx < OUTF * (KPANEL / 4); idx += 256) {
      const int n  = idx >> 4;
      const int kc = (idx & 15) * 4;
      const float4 wv = ((const float4*)W)[n * 64 + (kp >> 2) + (idx & 15)];
      ((float4*)(Ws + (size_t)n * WSTRIDE))[kc >> 2] = wv;
    }
    __syncthreads();

    const float* xrow = Xs + (size_t)(mt * 16 + lm) * XSTRIDE + kp;
    for (int k = 0; k < KPANEL; k += 4) {
      // A 16x4 fp32: lanes 0-15 hold {k0,k1}, lanes 16-31 hold {k2,k3}
      const v2f a = *(const v2f*)(xrow + k + 2 * half);
#pragma unroll
      for (int nt = 0; nt < 8; ++nt) {
        const int n = (ntBase + nt) * 16 + lm;
        // B 4x16 fp32, mirrored layout: one aligned b64 per lane
        const v2f bv = *(const v2f*)(Ws + (size_t)n * WSTRIDE + k + 2 * half);
        acc[nt] = __builtin_amdgcn_wmma_f32_16x16x4_f32(
            false, a, false, bv, (short)0, acc[nt], false, false);
      }
    }
  }

  // ---- epilogue: bias + residual, write o ----
#pragma unroll
  for (int nt = 0; nt < 8; ++nt) {
    const int n  = (ntBase + nt) * 16 + lm;
    const float bz = bias[n];
#pragma unroll
    for (int r = 0; r < 8; ++r) {
      const int rloc = mt * 16 + half * 8 + r;   // C layout: VGPR r -> M=r / M=8+r
      const int m  = m0 + rloc;
      const int bb = m / LP;
      const int lp = m - bb * LP;
      const float res =
          inp[((size_t)bb * LEN + lp + PAST) * DIM_IN + (n & 63)];
      o[((size_t)bb * LP + lp) * OUTF + n] = acc[nt][r] + bz + res;
    }
  }
}

// =====================================================================
extern "C" void kernel_launch(void* const* d_in, const int* in_sizes, int n_in,
                              void* d_out, int out_size, void* d_ws,
                              size_t ws_size, hipStream_t stream)
{
  (void)in_sizes; (void)n_in; (void)out_size; (void)ws_size;
  const float* inp   = (const float*)d_in[0];
  const float* W_out = (const float*)d_in[1];
  const float* b_out = (const float*)d_in[2];
  // d_in[3] (masks) is statically known: diff in {0, d}; unused.

  float* o_out = (float*)d_out;                                // B*LP*OUTF
  float* attn  = (float*)d_out + (size_t)BATCH * LP * OUTF;    // B*3*L*L
  float* out3  = (float*)d_ws;                                 // B*L*256 f32

  dim3 g1(LEN / TILE, BATCH);
  const size_t lds1 = (size_t)2 * BROWS * DIM_IN * sizeof(float);  // 147456 B
  attn_layers_kernel<<<g1, 256, lds1, stream>>>(inp, attn, out3);

  const int mblocks = (BATCH * LP) / PROJ_ROWS;                // 252
  const size_t lds2 =
      (size_t)(PROJ_ROWS * XSTRIDE + OUTF * WSTRIDE) * sizeof(float);
  proj_kernel<<<mblocks, 256, lds2, stream>>>(out3, W_out, b_out, inp, o_out);
}